// WarpformerLayer_80633716015215
// MI455X (gfx1250) — compile-verified
//
#include <hip/hip_runtime.h>
#include <hip/hip_bf16.h>
#include <math.h>

// ---------------- problem constants ----------------
#define B_    32
#define K_    23
#define L_    1024
#define D_    64
#define NEWL  48
#define NH_   4
#define DK_   16
#define DINNER 128
#define NLAYERS 3
#define LN_EPS 1e-5f

#define NSEQ  (B_ * K_)           // 736

// d_out layout (floats): z0 | new_pad_mask | almat
#define Z0_ELEMS   ((size_t)B_ * K_ * NEWL * D_)       // 2,260,992
#define PM_ELEMS   ((size_t)B_ * K_ * NEWL)            // 35,328
#define PM_OFF     (Z0_ELEMS)
#define ALMAT_OFF  (Z0_ELEMS + PM_ELEMS)

// LDS strides (padded against bank conflicts)
#define XPAD 72
#define SPAD 50
#define HPAD 132

typedef float v2f __attribute__((ext_vector_type(2)));
typedef float v8f __attribute__((ext_vector_type(8)));

// ----------------------------------------------------------------------------
// Wave-level 16x16 f32 WMMA tile:  acc += A(16xK, row-major lda) * B(KxN=16)
// B element (k,n) is B[k*sbk + n*sbn]  (sbn=ld for transposed access).
// A 16x4 frag layout (ISA 7.12.2): lanes 0-15: M=lane, K={k0,k0+1};
//                                  lanes 16-31: M=lane-16, K={k0+2,k0+3}.
// B 4x16 frag mirrors A; C/D: VGPR r -> row r (lanes 0-15) / r+8 (lanes 16-31).
// ----------------------------------------------------------------------------
__device__ __forceinline__ v8f wmma_tile(const float* A, int lda,
                                         const float* Bp, int sbk, int sbn,
                                         int K, v8f acc, int lane) {
  const int m  = lane & 15;
  const int kh = (lane >> 4) << 1;   // 0 or 2
  for (int k0 = 0; k0 < K; k0 += 4) {
    v2f a, b;
    a.x = A[m * lda + (k0 + kh)];
    a.y = A[m * lda + (k0 + kh + 1)];
    b.x = Bp[(k0 + kh) * sbk + m * sbn];
    b.y = Bp[(k0 + kh + 1) * sbk + m * sbn];
    acc = __builtin_amdgcn_wmma_f32_16x16x4_f32(false, a, false, b,
                                                (short)0, acc, false, false);
  }
  return acc;
}

__device__ __forceinline__ float gelu_tanh(float x) {
  float x3 = x * x * x;
  return 0.5f * x * (1.0f + tanhf(0.79788456080286535588f * (x + 0.044715f * x3)));
}

// ---------------------------------------------------------------------------
// K0: bucket ids + clamped per-bucket counts
// ---------------------------------------------------------------------------
__global__ void bucket_kernel(const float* __restrict__ et,
                              int* __restrict__ bucket,
                              float* __restrict__ total) {
  __shared__ int bins[NEWL];
  const int b = blockIdx.x, tid = threadIdx.x;
  if (tid < NEWL) bins[tid] = 0;
  __syncthreads();
  for (int l = tid; l < L_; l += blockDim.x) {
    float t = et[b * L_ + l];
    int bk = (t >= 0.0f && t < (float)NEWL) ? (int)floorf(t) : NEWL;
    bucket[b * L_ + l] = bk;
    if (bk < NEWL) atomicAdd(&bins[bk], 1);
  }
  __syncthreads();
  if (tid < NEWL) total[b * NEWL + tid] = fmaxf((float)bins[tid], 1.0f);
}

// ---------------------------------------------------------------------------
// K1: almat (B,K,L,NEWL) one-hot write, one float4 per thread (coalesced)
// ---------------------------------------------------------------------------
__global__ void almat_kernel(const int* __restrict__ bucket,
                             float* __restrict__ almat) {
  const long long idx = (long long)blockIdx.x * blockDim.x + threadIdx.x;
  const long long tot = (long long)B_ * K_ * L_ * (NEWL / 4);
  if (idx >= tot) return;
  int s4   = (int)(idx % (NEWL / 4));
  long long r = idx / (NEWL / 4);
  int l    = (int)(r % L_);
  long long r2 = r / L_;
  int k    = (int)(r2 % K_);
  int b    = (int)(r2 / K_);
  int bk   = bucket[b * L_ + l];
  float4 v;
  v.x = (bk == s4 * 4 + 0) ? 1.0f : 0.0f;
  v.y = (bk == s4 * 4 + 1) ? 1.0f : 0.0f;
  v.z = (bk == s4 * 4 + 2) ? 1.0f : 0.0f;
  v.w = (bk == s4 * 4 + 3) ? 1.0f : 0.0f;
  size_t off = ((((size_t)(b * K_ + k)) * L_ + l) * NEWL) + (size_t)s4 * 4;
  *(float4*)(almat + off) = v;
}

// ---------------------------------------------------------------------------
// K2: new_pad_mask (B,K,NEWL) = binned sum of non_pad_mask / total
// ---------------------------------------------------------------------------
__global__ void padmask_kernel(const float* __restrict__ mask,
                               const int* __restrict__ bucket,
                               const float* __restrict__ total,
                               float* __restrict__ pm) {
  __shared__ float bins[NEWL];
  const int n = blockIdx.x, b = n / K_, tid = threadIdx.x;
  if (tid < NEWL) bins[tid] = 0.0f;
  __syncthreads();
  for (int l = tid; l < L_; l += blockDim.x) {
    int bk = bucket[b * L_ + l];
    if (bk < NEWL) atomicAdd(&bins[bk], mask[(size_t)n * L_ + l]);
  }
  __syncthreads();
  if (tid < NEWL) pm[n * NEWL + tid] = bins[tid] / total[b * NEWL + tid];
}

// ---------------------------------------------------------------------------
// K3: segment max of h0 per bucket, clamped at 0 -> z0 region of d_out.
// Streams 193MB; coalesced across d. Running max + flush on bucket change,
// flushed with unsigned atomicMax (values >=0 -> uint order == float order).
// ---------------------------------------------------------------------------
__global__ __launch_bounds__(256) void segmax_kernel(
    const float* __restrict__ h0, const int* __restrict__ bucket,
    float* __restrict__ z0) {
  __shared__ int sb[L_];
  __shared__ unsigned bins[NEWL * D_];
  const int n = blockIdx.x;          // b*K + k
  const int b = n / K_;
  const int tid = threadIdx.x;
  for (int i = tid; i < L_; i += 256) sb[i] = bucket[b * L_ + i];
  for (int i = tid; i < NEWL * D_; i += 256) bins[i] = 0u;  // bits of +0.0f
  __syncthreads();
  const int d = tid & 63;
  const int q = tid >> 6;            // 0..3, quarter of l range
  const float* src = h0 + (size_t)n * L_ * D_;
  int cur = -1;
  float run = 0.0f;
  for (int l = q * 256; l < q * 256 + 256; ++l) {
    int bk = sb[l];
    float v = src[(size_t)l * D_ + d];
    if (bk != cur) {
      if (cur >= 0 && cur < NEWL)
        atomicMax(&bins[cur * D_ + d], __float_as_uint(fmaxf(run, 0.0f)));
      cur = bk;
      run = v;
    } else {
      run = fmaxf(run, v);
    }
  }
  if (cur >= 0 && cur < NEWL)
    atomicMax(&bins[cur * D_ + d], __float_as_uint(fmaxf(run, 0.0f)));
  __syncthreads();
  float* dst = z0 + (size_t)n * NEWL * D_;
  for (int i = tid; i < NEWL * D_; i += 256) dst[i] = __uint_as_float(bins[i]);
}

// ---------------------------------------------------------------------------
// K4: 3-layer transformer encoder, one block per sequence (48 x 64), all
// GEMMs via v_wmma_f32_16x16x4_f32. Dynamic LDS (~105KB), 8 waves.
// ---------------------------------------------------------------------------
__global__ __launch_bounds__(256) void encoder_kernel(
    float* __restrict__ z0, const float* __restrict__ pm,
    const float* __restrict__ Wq, const float* __restrict__ Wk,
    const float* __restrict__ Wv, const float* __restrict__ Wo,
    const float* __restrict__ w1, const float* __restrict__ b1,
    const float* __restrict__ w2, const float* __restrict__ b2,
    const float* __restrict__ g1, const float* __restrict__ be1,
    const float* __restrict__ g2, const float* __restrict__ be2) {
  extern __shared__ float smem[];
  float* XA  = smem;                  // 48*72  current x
  float* XB  = XA + 48 * XPAD;        // 48*72  residual+proj scratch
  float* Qs  = XB + 48 * XPAD;        // 48*72  Q, later ctx
  float* Ks  = Qs + 48 * XPAD;        // 48*72
  float* Vs  = Ks + 48 * XPAD;        // 48*72
  float* Ss  = Vs + 48 * XPAD;        // 4*48*50 scores; reused as H1 (48*132)
  float* H1  = Ss;
  float* mrow = Ss + 4 * 48 * SPAD;   // 48

  const int n = blockIdx.x, tid = threadIdx.x;
  const int lane = tid & 31, wave = tid >> 5;
  float* xg = z0 + (size_t)n * NEWL * D_;

  for (int i = tid; i < NEWL * D_; i += 256)
    XA[(i >> 6) * XPAD + (i & 63)] = xg[i];
  if (tid < NEWL) mrow[tid] = pm[n * NEWL + tid];
  __syncthreads();

  const v8f ZERO = {0.f, 0.f, 0.f, 0.f, 0.f, 0.f, 0.f, 0.f};

  for (int layer = 0; layer < NLAYERS; ++layer) {
    const float* wq  = Wq + layer * D_ * D_;
    const float* wk  = Wk + layer * D_ * D_;
    const float* wv  = Wv + layer * D_ * D_;
    const float* wo  = Wo + layer * D_ * D_;
    const float* w1l = w1 + layer * D_ * DINNER;
    const float* b1l = b1 + layer * DINNER;
    const float* w2l = w2 + layer * DINNER * D_;
    const float* b2l = b2 + layer * D_;
    const float* g1l = g1 + layer * D_;
    const float* e1l = be1 + layer * D_;
    const float* g2l = g2 + layer * D_;
    const float* e2l = be2 + layer * D_;

    // ---- QKV projections: 3 GEMMs of 48x64x64 -> 36 tiles ----
    for (int t = wave; t < 36; t += 8) {
      int which = t / 12, tt = t % 12, mi = tt >> 2, ni = tt & 3;
      const float* W = (which == 0) ? wq : (which == 1) ? wk : wv;
      float* O = (which == 0) ? Qs : (which == 1) ? Ks : Vs;
      v8f acc = wmma_tile(&XA[mi * 16 * XPAD], XPAD, W + ni * 16, D_, 1, D_,
                          ZERO, lane);
      int nn = lane & 15, mh = (lane >> 4) << 3;
#pragma unroll
      for (int r = 0; r < 8; ++r)
        O[(mi * 16 + mh + r) * XPAD + ni * 16 + nn] = acc[r];
    }
    __syncthreads();

    // ---- scores S_h = (Q_h @ K_h^T) * 0.25 : 4 heads x 3x3 tiles ----
    for (int t = wave; t < 36; t += 8) {
      int h = t / 9, tt = t % 9, mi = tt / 3, ni = tt % 3;
      v8f acc = wmma_tile(&Qs[mi * 16 * XPAD + h * 16], XPAD,
                          &Ks[ni * 16 * XPAD + h * 16], 1, XPAD, DK_,
                          ZERO, lane);
      int nn = lane & 15, mh = (lane >> 4) << 3;
      float* Sh = &Ss[h * 48 * SPAD];
#pragma unroll
      for (int r = 0; r < 8; ++r)
        Sh[(mi * 16 + mh + r) * SPAD + ni * 16 + nn] = acc[r] * 0.25f;
    }
    __syncthreads();

    // ---- softmax over keys: 4*48 = 192 rows, thread-per-row ----
    if (tid < NH_ * NEWL) {
      float* row = &Ss[(tid / NEWL) * 48 * SPAD + (tid % NEWL) * SPAD];
      float mx = -1e30f;
      for (int j = 0; j < NEWL; ++j) mx = fmaxf(mx, row[j]);
      float sum = 0.0f;
      for (int j = 0; j < NEWL; ++j) { float e = expf(row[j] - mx); row[j] = e; sum += e; }
      float inv = 1.0f / sum;
      for (int j = 0; j < NEWL; ++j) row[j] *= inv;
    }
    __syncthreads();

    // ---- ctx_h = S_h @ V_h : 4 heads x 3 tiles -> into Qs (Q dead) ----
    for (int t = wave; t < 12; t += 8) {
      int h = t / 3, mi = t % 3;
      v8f acc = wmma_tile(&Ss[h * 48 * SPAD + mi * 16 * SPAD], SPAD,
                          &Vs[h * 16], XPAD, 1, NEWL, ZERO, lane);
      int nn = lane & 15, mh = (lane >> 4) << 3;
#pragma unroll
      for (int r = 0; r < 8; ++r)
        Qs[(mi * 16 + mh + r) * XPAD + h * 16 + nn] = acc[r];
    }
    __syncthreads();

    // ---- XB = XA + ctx @ Wo : 12 tiles ----
    for (int t = wave; t < 12; t += 8) {
      int mi = t >> 2, ni = t & 3;
      v8f acc = wmma_tile(&Qs[mi * 16 * XPAD], XPAD, wo + ni * 16, D_, 1, D_,
                          ZERO, lane);
      int nn = lane & 15, mh = (lane >> 4) << 3;
#pragma unroll
      for (int r = 0; r < 8; ++r) {
        int m = mi * 16 + mh + r, c = ni * 16 + nn;
        XB[m * XPAD + c] = XA[m * XPAD + c] + acc[r];
      }
    }
    __syncthreads();

    // ---- LN1 (masked) -> XA ----
    if (tid < NEWL) {
      float* src = &XB[tid * XPAD];
      float mean = 0.0f;
      for (int c = 0; c < D_; ++c) mean += src[c];
      mean *= (1.0f / D_);
      float var = 0.0f;
      for (int c = 0; c < D_; ++c) { float d0 = src[c] - mean; var += d0 * d0; }
      var *= (1.0f / D_);
      float inv = rsqrtf(var + LN_EPS);
      float mk = mrow[tid];
      for (int c = 0; c < D_; ++c)
        XA[tid * XPAD + c] = ((src[c] - mean) * inv * g1l[c] + e1l[c]) * mk;
    }
    __syncthreads();

    // ---- H1 = gelu(XA @ w1 + b1): 3x8 tiles, K=64 ----
    for (int t = wave; t < 24; t += 8) {
      int mi = t >> 3, ni = t & 7;
      v8f acc = wmma_tile(&XA[mi * 16 * XPAD], XPAD, w1l + ni * 16, DINNER, 1,
                          D_, ZERO, lane);
      int nn = lane & 15, mh = (lane >> 4) << 3;
      float bb = b1l[ni * 16 + nn];
#pragma unroll
      for (int r = 0; r < 8; ++r)
        H1[(mi * 16 + mh + r) * HPAD + ni * 16 + nn] = gelu_tanh(acc[r] + bb);
    }
    __syncthreads();

    // ---- XB = XA + H1 @ w2 + b2 : 12 tiles, K=128 ----
    for (int t = wave; t < 12; t += 8) {
      int mi = t >> 2, ni = t & 3;
      v8f acc = wmma_tile(&H1[mi * 16 * HPAD], HPAD, w2l + ni * 16, D_, 1,
                          DINNER, ZERO, lane);
      int nn = lane & 15, mh = (lane >> 4) << 3;
      float bb = b2l[ni * 16 + nn];
#pragma unroll
      for (int r = 0; r < 8; ++r) {
        int m = mi * 16 + mh + r, c = ni * 16 + nn;
        XB[m * XPAD + c] = XA[m * XPAD + c] + acc[r] + bb;
      }
    }
    __syncthreads();

    // ---- LN2 (masked) -> XA ----
    if (tid < NEWL) {
      float* src = &XB[tid * XPAD];
      float mean = 0.0f;
      for (int c = 0; c < D_; ++c) mean += src[c];
      mean *= (1.0f / D_);
      float var = 0.0f;
      for (int c = 0; c < D_; ++c) { float d0 = src[c] - mean; var += d0 * d0; }
      var *= (1.0f / D_);
      float inv = rsqrtf(var + LN_EPS);
      float mk = mrow[tid];
      for (int c = 0; c < D_; ++c)
        XA[tid * XPAD + c] = ((src[c] - mean) * inv * g2l[c] + e2l[c]) * mk;
    }
    __syncthreads();
  }

  for (int i = tid; i < NEWL * D_; i += 256)
    xg[i] = XA[(i >> 6) * XPAD + (i & 63)];
}

// ---------------------------------------------------------------------------
extern "C" void kernel_launch(void* const* d_in, const int* in_sizes, int n_in,
                              void* d_out, int out_size, void* d_ws, size_t ws_size,
                              hipStream_t stream) {
  const float* h0  = (const float*)d_in[0];
  const float* npm = (const float*)d_in[1];
  const float* et  = (const float*)d_in[2];
  const float* Wq  = (const float*)d_in[3];
  const float* Wk  = (const float*)d_in[4];
  const float* Wv  = (const float*)d_in[5];
  const float* Wo  = (const float*)d_in[6];
  const float* w1  = (const float*)d_in[7];
  const float* b1  = (const float*)d_in[8];
  const float* w2  = (const float*)d_in[9];
  const float* b2  = (const float*)d_in[10];
  const float* g1  = (const float*)d_in[11];
  const float* be1 = (const float*)d_in[12];
  const float* g2  = (const float*)d_in[13];
  const float* be2 = (const float*)d_in[14];

  float* out = (float*)d_out;
  int*   bucket = (int*)d_ws;
  float* total  = (float*)((char*)d_ws + (size_t)B_ * L_ * sizeof(int));

  // K0: buckets + totals
  bucket_kernel<<<B_, 256, 0, stream>>>(et, bucket, total);

  // K1: almat (one float4 per thread)
  {
    long long tot = (long long)B_ * K_ * L_ * (NEWL / 4);
    int blocks = (int)((tot + 255) / 256);
    almat_kernel<<<blocks, 256, 0, stream>>>(bucket, out + ALMAT_OFF);
  }

  // K2: new_pad_mask
  padmask_kernel<<<NSEQ, 256, 0, stream>>>(npm, bucket, total, out + PM_OFF);

  // K3: segment max -> z0 region
  segmax_kernel<<<NSEQ, 256, 0, stream>>>(h0, bucket, out);

  // K4: 3-layer encoder in-place on z0 region (WMMA f32)
  {
    size_t smem = (size_t)(5 * 48 * XPAD + 4 * 48 * SPAD + 48) * sizeof(float);
    (void)hipFuncSetAttribute((const void*)encoder_kernel,
                              hipFuncAttributeMaxDynamicSharedMemorySize,
                              (int)smem);
    encoder_kernel<<<NSEQ, 256, smem, stream>>>(
        out, out + PM_OFF, Wq, Wk, Wv, Wo, w1, b1, w2, b2, g1, be1, g2, be2);
  }
}